// BiDGNBlock_8770323218991
// MI455X (gfx1250) — compile-verified
//
#include <hip/hip_runtime.h>

// BiDGN block for MI455X (gfx1250): fp32 WMMA (V_WMMA_F32_16X16X4_F32) for all
// dense GEMMs (exact fp32 — the problem is latency/BW bound, ~0.7 GFLOP /
// ~20MB, so low-precision WMMA buys nothing and costs accuracy).
// Round 2 -> 3: templated QKV (no branches in inner loops), split attention
// side-loops, explicit b64 vector loads, unroll-by-4 K loops.

#define B_   8
#define C_   64
#define T_   256
#define R_   (B_ * C_)   // 512 rows = B*C
#define EXP_ 32

typedef float v2f __attribute__((ext_vector_type(2)));
typedef float v8f __attribute__((ext_vector_type(8)));

// Only check the builtin in the DEVICE pass: the host x86 pass has no amdgcn
// builtins and would trip the #error spuriously.
#if defined(__HIP_DEVICE_COMPILE__) && \
    !__has_builtin(__builtin_amdgcn_wmma_f32_16x16x4_f32)
#error "gfx1250 wmma f32 16x16x4 builtin not available on this toolchain"
#endif

__device__ __forceinline__ v8f wmma4(v2f a, v2f b, v8f c) {
#if defined(__HIP_DEVICE_COMPILE__)
  // (neg_a, A, neg_b, B, c_mod, C, reuse_a, reuse_b)
  return __builtin_amdgcn_wmma_f32_16x16x4_f32(false, a, false, b, (short)0, c,
                                               false, false);
#else
  (void)a; (void)b;
  return c;
#endif
}

// 8-byte aligned pair load (k is always even; rows are 1KB aligned).
__device__ __forceinline__ v2f ld2(const float* __restrict__ p) {
  return *(const v2f*)p;
}

// ---------------------------------------------------------------------------
// K1 (x3 template instantiations): O = A_src @ W^T + bias
//   WHICH=0: A=x_l (->Q)   WHICH=1: A=x_r (->K)   WHICH=2: A=x_l-x_r (->V)
// One wave per 16x16 output tile. grid = (R/16, T/16), block = 32.
// ---------------------------------------------------------------------------
template <int WHICH>
__global__ __launch_bounds__(32) void qkv_kernel(
    const float* __restrict__ xl, const float* __restrict__ xr,
    const float* __restrict__ W, const float* __restrict__ bb,
    float* __restrict__ O)
{
  const int lane = threadIdx.x & 31;
  const int m0 = blockIdx.x * 16;   // over R_
  const int n0 = blockIdx.y * 16;   // over T_

  const int rA  = m0 + (lane & 15);
  const int cB  = n0 + (lane & 15);
  const int khi = (lane >> 4) * 2;

  const float* a0 = (WHICH == 1 ? xr : xl) + rA * T_ + khi;
  const float* a1 = xr + rA * T_ + khi;          // only used when WHICH==2
  const float* wp = W + cB * T_ + khi;

  v8f acc = {};
#pragma unroll 4
  for (int kb = 0; kb < T_; kb += 4) {
    v2f a;
    if (WHICH == 2) {
      const v2f u = ld2(a0 + kb);
      const v2f w = ld2(a1 + kb);
      a.x = u.x - w.x;
      a.y = u.y - w.y;
    } else {
      a = ld2(a0 + kb);
    }
    const v2f b = ld2(wp + kb);   // B[k][col] = W[col][k]  (X @ W^T)
    acc = wmma4(a, b, acc);
  }
  const float bias = bb[cB];
  const int rb = m0 + (lane >> 4) * 8;
#pragma unroll
  for (int r = 0; r < 8; ++r)
    O[(rb + r) * T_ + cB] = acc[r] + bias;
}

// ---------------------------------------------------------------------------
// K2: per-batch attention. energy = QK^T/16 -> softmax -> Ol = A@V, Or = A^T@V
// grid = B, block = 256 (8 waves). attn matrix lives in 16KB static LDS.
// ---------------------------------------------------------------------------
__global__ __launch_bounds__(256) void attn_kernel(
    const float* __restrict__ Q, const float* __restrict__ K,
    const float* __restrict__ V,
    float* __restrict__ Ol, float* __restrict__ Or)
{
  __shared__ float As[C_ * C_];
  const int b    = blockIdx.x;
  const int tid  = threadIdx.x;
  const int wave = tid >> 5, lane = tid & 31;
  const int base = b * C_ * T_;
  const float* Qb = Q + base;
  const float* Kb = K + base;
  const float* Vb = V + base;
  const int khi = (lane >> 4) * 2;

  // ---- energy: 4x4 tiles of 16x16, K=256; 16 tiles over 8 waves ----
  for (int t = wave; t < 16; t += 8) {
    const int ti = t >> 2, tj = t & 3;
    const float* qp = Qb + (ti * 16 + (lane & 15)) * T_ + khi;
    const float* kp = Kb + (tj * 16 + (lane & 15)) * T_ + khi;
    v8f acc = {};
#pragma unroll 4
    for (int kb = 0; kb < T_; kb += 4)
      acc = wmma4(ld2(qp + kb), ld2(kp + kb), acc);
    const int rb = ti * 16 + (lane >> 4) * 8;
    const int cC = tj * 16 + (lane & 15);
#pragma unroll
    for (int r = 0; r < 8; ++r)
      As[(rb + r) * C_ + cC] = acc[r] * 0.0625f;   // 1/sqrt(256)
  }
  __syncthreads();

  // ---- row softmax over 64 cols (threads 0..63) ----
  if (tid < C_) {
    float mx = -3.4e38f;
    for (int j = 0; j < C_; ++j) mx = fmaxf(mx, As[tid * C_ + j]);
    float sum = 0.f;
    for (int j = 0; j < C_; ++j) {
      const float e = __expf(As[tid * C_ + j] - mx);
      As[tid * C_ + j] = e;
      sum += e;
    }
    const float inv = 1.f / sum;
    for (int j = 0; j < C_; ++j) As[tid * C_ + j] *= inv;
  }
  __syncthreads();

  // ---- out_l = A@V : 64 tiles of 16x16, K=64 ----
  for (int t = wave; t < 64; t += 8) {
    const int ti = t >> 4, tj = t & 15;
    const int rA = ti * 16 + (lane & 15);
    const int cC = tj * 16 + (lane & 15);
    v8f acc = {};
#pragma unroll 4
    for (int kb = 0; kb < C_; kb += 4) {
      const int k = kb + khi;
      v2f a;
      a.x = As[rA * C_ + k];
      a.y = As[rA * C_ + k + 1];
      v2f bb;
      bb.x = Vb[k * T_ + cC];
      bb.y = Vb[(k + 1) * T_ + cC];
      acc = wmma4(a, bb, acc);
    }
    const int rb = ti * 16 + (lane >> 4) * 8;
#pragma unroll
    for (int r = 0; r < 8; ++r)
      Ol[base + (rb + r) * T_ + cC] = acc[r];
  }

  // ---- out_r = A^T@V : 64 tiles of 16x16, K=64 ----
  for (int t = wave; t < 64; t += 8) {
    const int ti = t >> 4, tj = t & 15;
    const int rA = ti * 16 + (lane & 15);
    const int cC = tj * 16 + (lane & 15);
    v8f acc = {};
#pragma unroll 4
    for (int kb = 0; kb < C_; kb += 4) {
      const int k = kb + khi;
      v2f a;
      a.x = As[k * C_ + rA];        // transposed read
      a.y = As[(k + 1) * C_ + rA];
      v2f bb;
      bb.x = Vb[k * T_ + cC];
      bb.y = Vb[(k + 1) * T_ + cC];
      acc = wmma4(a, bb, acc);
    }
    const int rb = ti * 16 + (lane >> 4) * 8;
#pragma unroll
    for (int r = 0; r < 8; ++r)
      Or[base + (rb + r) * T_ + cC] = acc[r];
  }
}

// ---------------------------------------------------------------------------
// K3: P = In @ Wp^T + bp for both sides. grid = (R/16, T/16, 2), block = 32.
// (side select hoisted to SGPR pointers before the loop — stays uniform)
// ---------------------------------------------------------------------------
__global__ __launch_bounds__(32) void proj_kernel(
    const float* __restrict__ Il, const float* __restrict__ Ir,
    const float* __restrict__ Wp, const float* __restrict__ bp,
    float* __restrict__ Pl, float* __restrict__ Pr)
{
  const int lane = threadIdx.x & 31;
  const int m0 = blockIdx.x * 16, n0 = blockIdx.y * 16;
  const float* In = blockIdx.z ? Ir : Il;
  float*       O  = blockIdx.z ? Pr : Pl;
  const int rA  = m0 + (lane & 15);
  const int cB  = n0 + (lane & 15);
  const int khi = (lane >> 4) * 2;

  const float* ip = In + rA * T_ + khi;
  const float* wp = Wp + cB * T_ + khi;

  v8f acc = {};
#pragma unroll 4
  for (int kb = 0; kb < T_; kb += 4)
    acc = wmma4(ld2(ip + kb), ld2(wp + kb), acc);

  const float bias = bp[cB];
  const int rb = m0 + (lane >> 4) * 8;
#pragma unroll
  for (int r = 0; r < 8; ++r)
    O[(rb + r) * T_ + cB] = acc[r] + bias;
}

// ---------------------------------------------------------------------------
// K4/K7: O = LayerNorm(X)*g + b + res   (eps=1e-5). One block per row of 256.
// ---------------------------------------------------------------------------
__global__ __launch_bounds__(256) void ln_add_kernel(
    const float* __restrict__ X, const float* __restrict__ g,
    const float* __restrict__ bta, const float* __restrict__ res,
    float* __restrict__ O)
{
  __shared__ float red[256];
  __shared__ float stat[2];
  const int row = blockIdx.x, tid = threadIdx.x;
  const float v = X[row * T_ + tid];
  red[tid] = v;
  __syncthreads();
  for (int s = 128; s > 0; s >>= 1) {
    if (tid < s) red[tid] += red[tid + s];
    __syncthreads();
  }
  if (tid == 0) stat[0] = red[0] * (1.0f / T_);
  __syncthreads();
  const float m = stat[0];
  const float d = v - m;
  red[tid] = d * d;
  __syncthreads();
  for (int s = 128; s > 0; s >>= 1) {
    if (tid < s) red[tid] += red[tid + s];
    __syncthreads();
  }
  if (tid == 0) stat[1] = rsqrtf(red[0] * (1.0f / T_) + 1e-5f);
  __syncthreads();
  O[row * T_ + tid] = d * stat[1] * g[tid] + bta[tid] + res[row * T_ + tid];
}

// ---------------------------------------------------------------------------
// K5: cosine router. xp = concat(Al,Ar) @ Wr^T + br; normalize; cosine sims
// vs normalized centers; top-2 indices; s = sum(softmax(top2)) (== 1).
// One block per (b,c) row. block = 256.
// ---------------------------------------------------------------------------
__global__ __launch_bounds__(256) void router_kernel(
    const float* __restrict__ Al, const float* __restrict__ Ar,
    const float* __restrict__ centers, const float* __restrict__ Wr,
    const float* __restrict__ br, int* __restrict__ topi,
    float* __restrict__ sv)
{
  __shared__ float xcs[2 * T_];
  __shared__ float red[256];
  __shared__ float xps[EXP_];
  __shared__ float sims[C_];
  __shared__ float invn_s;
  const int row = blockIdx.x, tid = threadIdx.x;
  xcs[tid]      = Al[row * T_ + tid];
  xcs[T_ + tid] = Ar[row * T_ + tid];
  __syncthreads();

  // xp[j] = dot(xc, Wr[j]) + br[j]; 8 threads per j
  const int j = tid >> 3, sub = tid & 7;
  float p = 0.f;
  for (int d = sub; d < 2 * T_; d += 8) p += xcs[d] * Wr[j * (2 * T_) + d];
  red[tid] = p;
  __syncthreads();
  if (sub < 4) red[tid] += red[tid + 4];
  __syncthreads();
  if (sub < 2) red[tid] += red[tid + 2];
  __syncthreads();
  if (sub == 0) xps[j] = red[tid] + red[tid + 1] + br[j];
  __syncthreads();

  if (tid == 0) {
    float s2 = 0.f;
    for (int i = 0; i < EXP_; ++i) s2 += xps[i] * xps[i];
    invn_s = 1.f / fmaxf(sqrtf(s2), 1e-12f);
  }
  __syncthreads();

  if (tid < C_) {
    float cn = 0.f, dp = 0.f;
    for (int d = 0; d < EXP_; ++d) {
      const float cv = centers[tid * EXP_ + d];
      cn += cv * cv;
      dp += cv * xps[d];
    }
    sims[tid] = dp * invn_s / fmaxf(sqrtf(cn), 1e-12f);
  }
  __syncthreads();

  if (tid == 0) {
    int i1 = 0; float v1 = sims[0];
    for (int e = 1; e < C_; ++e)
      if (sims[e] > v1) { v1 = sims[e]; i1 = e; }
    int i2 = (i1 == 0) ? 1 : 0; float v2 = sims[i2];
    for (int e = 0; e < C_; ++e)
      if (e != i1 && sims[e] > v2) { v2 = sims[e]; i2 = e; }
    const float mx = fmaxf(v1, v2);
    const float e1 = __expf(v1 - mx), e2 = __expf(v2 - mx);
    sv[row] = (e1 + e2) / (e1 + e2);   // softmax prob-sum (analytically 1)
    topi[row * 2]     = i1;
    topi[row * 2 + 1] = i2;
  }
}

// ---------------------------------------------------------------------------
// K6: per-row 2-expert matvec: M[row] = s * sum_k ( We[e_k] @ a_row + be[e_k] )
// grid = (R, 2 sides), block = 256 (one output element per thread).
// ---------------------------------------------------------------------------
__global__ __launch_bounds__(256) void moe_kernel(
    const float* __restrict__ Al, const float* __restrict__ Ar,
    const float* __restrict__ We, const float* __restrict__ be,
    const int* __restrict__ topi, const float* __restrict__ sv,
    float* __restrict__ Ml, float* __restrict__ Mr)
{
  __shared__ float xs[T_];
  const int row = blockIdx.x, tid = threadIdx.x;
  const float* A = blockIdx.y ? Ar : Al;
  float*       M = blockIdx.y ? Mr : Ml;
  xs[tid] = A[row * T_ + tid];
  __syncthreads();
  const int e0 = topi[row * 2], e1 = topi[row * 2 + 1];
  const float s = sv[row];
  const float* W0 = We + (size_t)e0 * T_ * T_ + (size_t)tid * T_;
  const float* W1 = We + (size_t)e1 * T_ * T_ + (size_t)tid * T_;
  float acc = be[e0 * T_ + tid] + be[e1 * T_ + tid];
  for (int t = 0; t < T_; ++t) acc += (W0[t] + W1[t]) * xs[t];
  M[row * T_ + tid] = s * acc;
}

// ---------------------------------------------------------------------------
extern "C" void kernel_launch(void* const* d_in, const int* in_sizes, int n_in,
                              void* d_out, int out_size, void* d_ws,
                              size_t ws_size, hipStream_t stream)
{
  (void)in_sizes; (void)n_in; (void)out_size; (void)ws_size;

  const float* xl      = (const float*)d_in[0];
  const float* xr      = (const float*)d_in[1];
  const float* Wq      = (const float*)d_in[2];
  const float* bq      = (const float*)d_in[3];
  const float* Wk      = (const float*)d_in[4];
  const float* bk      = (const float*)d_in[5];
  const float* Wv      = (const float*)d_in[6];
  const float* bv      = (const float*)d_in[7];
  const float* Wp      = (const float*)d_in[8];
  const float* bp      = (const float*)d_in[9];
  const float* agl     = (const float*)d_in[10];
  const float* abl     = (const float*)d_in[11];
  const float* agr     = (const float*)d_in[12];
  const float* abr     = (const float*)d_in[13];
  const float* centers = (const float*)d_in[14];
  const float* Wr      = (const float*)d_in[15];
  const float* br      = (const float*)d_in[16];
  const float* We      = (const float*)d_in[17];
  const float* be      = (const float*)d_in[18];
  const float* mgl     = (const float*)d_in[19];
  const float* mbl     = (const float*)d_in[20];
  const float* mgr     = (const float*)d_in[21];
  const float* mbr     = (const float*)d_in[22];

  const int RT = R_ * T_;   // 131072 floats per (B,C,T) tensor
  float* f  = (float*)d_ws;
  float* Q  = f + 0 * RT;
  float* Kb = f + 1 * RT;
  float* V  = f + 2 * RT;
  float* Pr = f + 3 * RT;
  float* Al = f + 4 * RT;
  float* Ar = f + 5 * RT;
  float* Ol = f + 6 * RT;
  float* Or = f + 7 * RT;
  float* Pl = V;            // V is dead after attention
  float* Ml = Q;            // Q/K dead after attention+proj
  float* Mr = Kb;
  int*   topi = (int*)(f + 8 * RT);
  float* sv   = f + 8 * RT + 2 * R_;

  float* out_l2 = (float*)d_out;
  float* out_r2 = out_l2 + RT;

  const dim3 gTiles(R_ / 16, T_ / 16);

  // Stage 1: bi-attention
  qkv_kernel<0><<<gTiles, 32, 0, stream>>>(xl, xr, Wq, bq, Q);
  qkv_kernel<1><<<gTiles, 32, 0, stream>>>(xl, xr, Wk, bk, Kb);
  qkv_kernel<2><<<gTiles, 32, 0, stream>>>(xl, xr, Wv, bv, V);
  attn_kernel<<<B_, 256, 0, stream>>>(Q, Kb, V, Ol, Or);
  proj_kernel<<<dim3(R_ / 16, T_ / 16, 2), 32, 0, stream>>>(
      Ol, Or, Wp, bp, Pl, Pr);
  ln_add_kernel<<<R_, 256, 0, stream>>>(Pl, agl, abl, xl, Al);
  ln_add_kernel<<<R_, 256, 0, stream>>>(Pr, agr, abr, xr, Ar);

  // Stage 2: SGMoE
  router_kernel<<<R_, 256, 0, stream>>>(Al, Ar, centers, Wr, br, topi, sv);
  moe_kernel<<<dim3(R_, 2), 256, 0, stream>>>(Al, Ar, We, be, topi, sv, Ml, Mr);
  ln_add_kernel<<<R_, 256, 0, stream>>>(Ml, mgl, mbl, Al, out_l2);
  ln_add_kernel<<<R_, 256, 0, stream>>>(Mr, mgr, mbr, Ar, out_r2);
}